// Head_33981781246191
// MI455X (gfx1250) — compile-verified
//
#include <hip/hip_runtime.h>
#include <hip/hip_bf16.h>

// ---------------------------------------------------------------------------
// Fused single-head attention with KV-cache extension, fp32 WMMA (CDNA5).
//   out  : (8,1024,64)        d_out[0 .. 524287]
//   k    : (8,4096,64)        d_out[524288 .. 2621439]
//   v    : (8,4096,64)        d_out[2621440 .. 4718591]
// Attention kernel double-buffers K/V tiles through LDS using CDNA5
// GLOBAL_LOAD_ASYNC_TO_LDS_B128 (ASYNCcnt) when the toolchain exposes it.
// ---------------------------------------------------------------------------

typedef __attribute__((ext_vector_type(2))) float v2f;
typedef __attribute__((ext_vector_type(8))) float v8f;
typedef __attribute__((ext_vector_type(4))) int   v4i;

// address-space-qualified pointee types for the async builtin
typedef __attribute__((address_space(1))) v4i gv4i;   // global
typedef __attribute__((address_space(3))) v4i lv4i;   // LDS

#define WMMA_F32(A, B, C) \
  __builtin_amdgcn_wmma_f32_16x16x4_f32(false, (A), false, (B), (short)0, (C), false, false)

static constexpr int B_   = 8;
static constexpr int TN   = 1024;
static constexpr int TC   = 3072;
static constexpr int TT   = 4096;   // TC + TN
static constexpr int HD   = 64;
static constexpr int ED   = 512;

// ----- CDNA5 async global->LDS helpers (guarded; fallback stays correct) ----
__device__ __forceinline__ void async_b128(const float* g, float* l) {
#if defined(__gfx1250__) && __has_builtin(__builtin_amdgcn_global_load_async_to_lds_b128)
  float* gnc = const_cast<float*>(g);
  __builtin_amdgcn_global_load_async_to_lds_b128(
      (gv4i*)gnc, (lv4i*)l, /*imm offset*/0, /*imm cpol*/0);
#else
  *(float4*)l = *(const float4*)g;
#endif
}

__device__ __forceinline__ void wait_async0() {
#if defined(__gfx1250__) && __has_builtin(__builtin_amdgcn_s_wait_asynccnt)
  __builtin_amdgcn_s_wait_asynccnt(0);
#endif
  asm volatile("" ::: "memory");          // keep LDS reads below the wait
}

// ---------------------------------------------------------------------------
// Kernel 1: copy cache_k / cache_v into rows [0,3072) of the output k / v.
// ---------------------------------------------------------------------------
__global__ void cache_copy_kernel(const float4* __restrict__ ck,
                                  const float4* __restrict__ cv,
                                  float4* __restrict__ kout,
                                  float4* __restrict__ vout) {
  int idx = blockIdx.x * 256 + threadIdx.x;           // 0 .. 393215
  const int per_b = TC * (HD / 4);                    // 49152 float4 per batch
  int b = idx / per_b, r = idx - b * per_b;
  int dst = b * (TT * (HD / 4)) + r;
  kout[dst] = ck[idx];
  vout[dst] = cv[idx];
}

// ---------------------------------------------------------------------------
// Kernel 2: projections via V_WMMA_F32_16X16X4_F32.
// One wave per (16-row block, matrix, 16-col tile): 512 * 3 * 4 = 6144 waves.
// ---------------------------------------------------------------------------
__global__ __launch_bounds__(128) void proj_kernel(
    const float* __restrict__ x,
    const float* __restrict__ Wq, const float* __restrict__ Wk,
    const float* __restrict__ Wv,
    float* __restrict__ qws, float* __restrict__ kout, float* __restrict__ vout) {
  const int lane  = threadIdx.x & 31;
  const int wid   = blockIdx.x * (blockDim.x >> 5) + (threadIdx.x >> 5);
  const int rowblk = wid / 12;          // 0..511  (16 rows of the 8192-row X)
  const int combo  = wid % 12;
  const int mat    = combo >> 2;        // 0=q, 1=k, 2=v
  const int nt     = combo & 3;         // 16-col tile of the 64-col output
  const float* W = (mat == 0) ? Wq : ((mat == 1) ? Wk : Wv);

  const int mloc = lane & 15;
  const int h    = lane >> 4;
  const int kb   = h << 1;              // A/B lane-half K offset
  const int arow = rowblk * 16 + mloc;  // flat row = b*1024 + t
  const int ncol = nt * 16 + mloc;

  v8f c = {};
  for (int k0 = 0; k0 < ED; k0 += 4) {
    v2f a, bm;
    a.x  = x[arow * ED + k0 + kb];
    a.y  = x[arow * ED + k0 + kb + 1];
    bm.x = W[(k0 + kb) * HD + ncol];
    bm.y = W[(k0 + kb + 1) * HD + ncol];
    c = WMMA_F32(a, bm, c);
  }

#pragma unroll
  for (int j = 0; j < 8; ++j) {
    int orow = rowblk * 16 + j + 8 * h;           // b*1024 + t
    float val = c[j];
    if (mat == 0) {
      qws[orow * HD + ncol] = val;
    } else {
      int b = orow >> 10, t = orow & 1023;
      float* dst = (mat == 1) ? kout : vout;
      dst[(b * TT + TC + t) * HD + ncol] = val;
    }
  }
}

// ---------------------------------------------------------------------------
// Kernel 3: flash attention. One wave per (batch, 16-query block) = 512 waves.
// Double-buffered async K/V staging; P transposed C->A layout through LDS.
// Per-wave LDS: 2x(K 16x68) + 2x(V 16x68) + P 16x17 = 4624 floats.
// ---------------------------------------------------------------------------
__global__ __launch_bounds__(64) void attn_kernel(
    const float* __restrict__ q,
    const float* __restrict__ kbuf, const float* __restrict__ vbuf,
    float* __restrict__ outbuf) {
  __shared__ float lds[2 * 4624];
  const int lane  = threadIdx.x & 31;
  const int wslot = threadIdx.x >> 5;
  float* Kl = lds + wslot * 4624;       // [parity * 1088]
  float* Vl = Kl + 2176;
  float* Pl = Vl + 2176;                // 272 floats

  const int wid = blockIdx.x * 2 + wslot;   // 0..511
  const int b   = wid >> 6;
  const int qb  = wid & 63;
  const int t0  = qb * 16;

  const int mloc = lane & 15;
  const int h    = lane >> 4;
  const int kb   = h << 1;

  const float* Kbase = kbuf + b * TT * HD;
  const float* Vbase = vbuf + b * TT * HD;

  // Preload Q tile (16x64) in WMMA A layout, pre-scaled by d^-0.5 = 0.125.
  const float* qrow = q + (b * TN + t0 + mloc) * HD;
  v2f qa[16];
#pragma unroll
  for (int i = 0; i < 16; ++i) {
    qa[i].x = qrow[4 * i + kb]     * 0.125f;
    qa[i].y = qrow[4 * i + kb + 1] * 0.125f;
  }

  v8f oacc[4] = {};
  float mrow[8], lrow[8];
#pragma unroll
  for (int j = 0; j < 8; ++j) { mrow[j] = -3.0e38f; lrow[j] = 0.0f; }

  // Issue one 16x64 K tile + V tile into LDS buffer `p` (16B per lane per op).
  auto issue_tile = [&](int st, int p) {
    const int s0 = st * 16;
    float* Kd = Kl + p * 1088;
    float* Vd = Vl + p * 1088;
#pragma unroll
    for (int it = 0; it < 8; ++it) {
      int idx = it * 32 + lane;             // 0..255 quads
      int r = idx >> 4, qd = (idx & 15) << 2;
      async_b128(Kbase + (s0 + r) * HD + qd, Kd + r * 68 + qd);
      async_b128(Vbase + (s0 + r) * HD + qd, Vd + r * 68 + qd);
    }
  };

  const int ntiles = (TC + t0) / 16 + 1;    // 193 + qb; last tile is diagonal

  issue_tile(0, 0);
  for (int st = 0; st < ntiles; ++st) {
    const int p = st & 1;
    wait_async0();                          // tile st resident in LDS
    __builtin_amdgcn_wave_barrier();

    if (st + 1 < ntiles) {
      // prior DS reads of the other buffer must be complete before overwrite
      asm volatile("s_wait_dscnt 0x0" ::: "memory");
      issue_tile(st + 1, (st + 1) & 1);
    }

    const float* Kp = Kl + p * 1088;
    const float* Vp = Vl + p * 1088;

    // scores = (Q/8) @ K^T : 16 fp32 WMMA steps over head dim
    v8f c = {};
#pragma unroll
    for (int i = 0; i < 16; ++i) {
      v2f bk;
      bk.x = Kp[mloc * 68 + 4 * i + kb];
      bk.y = Kp[mloc * 68 + 4 * i + kb + 1];
      c = WMMA_F32(qa[i], bk, c);
    }

    // causal mask on the diagonal tile: col n valid iff n <= row m
    if (st == ntiles - 1) {
#pragma unroll
      for (int j = 0; j < 8; ++j) {
        int mr = j + 8 * h;
        if (mloc > mr) c[j] = -3.0e38f;
      }
    }

    // online softmax (rows live in 16-lane groups)
    float alpha[8], pr[8];
#pragma unroll
    for (int j = 0; j < 8; ++j) {
      float v = c[j];
      v = fmaxf(v, __shfl_xor(v, 1, 16));
      v = fmaxf(v, __shfl_xor(v, 2, 16));
      v = fmaxf(v, __shfl_xor(v, 4, 16));
      v = fmaxf(v, __shfl_xor(v, 8, 16));
      float mnew = fmaxf(mrow[j], v);
      alpha[j] = __expf(mrow[j] - mnew);
      mrow[j]  = mnew;
      pr[j] = __expf(c[j] - mnew);
      float s = pr[j];
      s += __shfl_xor(s, 1, 16);
      s += __shfl_xor(s, 2, 16);
      s += __shfl_xor(s, 4, 16);
      s += __shfl_xor(s, 8, 16);
      lrow[j] = lrow[j] * alpha[j] + s;
    }

    // rescale accumulators (same row<->(vgpr,lane-half) mapping as C layout)
#pragma unroll
    for (int nt = 0; nt < 4; ++nt)
#pragma unroll
      for (int j = 0; j < 8; ++j) oacc[nt][j] *= alpha[j];

    // transpose P from C layout to A layout through LDS (stride 17)
#pragma unroll
    for (int j = 0; j < 8; ++j) Pl[(j + 8 * h) * 17 + mloc] = pr[j];
    __builtin_amdgcn_wave_barrier();

    // O += P @ Vtile : 4 K-chunks x 4 N-tiles of fp32 WMMA
#pragma unroll
    for (int s = 0; s < 4; ++s) {
      v2f pa;
      pa.x = Pl[mloc * 17 + 4 * s + kb];
      pa.y = Pl[mloc * 17 + 4 * s + kb + 1];
#pragma unroll
      for (int nt = 0; nt < 4; ++nt) {
        v2f bv;
        bv.x = Vp[(4 * s + kb) * 68 + nt * 16 + mloc];
        bv.y = Vp[(4 * s + kb + 1) * 68 + nt * 16 + mloc];
        oacc[nt] = WMMA_F32(pa, bv, oacc[nt]);
      }
    }
    __builtin_amdgcn_wave_barrier();
  }

  // epilogue: out = O / l
#pragma unroll
  for (int j = 0; j < 8; ++j) {
    float inv = 1.0f / lrow[j];
    int trow = t0 + j + 8 * h;
#pragma unroll
    for (int nt = 0; nt < 4; ++nt) {
      outbuf[(b * TN + trow) * HD + nt * 16 + mloc] = oacc[nt][j] * inv;
    }
  }
}

// ---------------------------------------------------------------------------
extern "C" void kernel_launch(void* const* d_in, const int* in_sizes, int n_in,
                              void* d_out, int out_size, void* d_ws, size_t ws_size,
                              hipStream_t stream) {
  const float* x  = (const float*)d_in[0];
  // d_in[1] = mask (int32); causal structure is computed analytically.
  const float* ck = (const float*)d_in[2];
  const float* cv = (const float*)d_in[3];
  const float* Wq = (const float*)d_in[4];
  const float* Wk = (const float*)d_in[5];
  const float* Wv = (const float*)d_in[6];

  float* out  = (float*)d_out;
  float* kout = out + (size_t)B_ * TN * HD;           // 524288
  float* vout = kout + (size_t)B_ * TT * HD;          // +2097152
  float* qws  = (float*)d_ws;                         // 8*1024*64 fp32 = 2 MB

  // 1) cache -> output k/v rows [0,3072)
  cache_copy_kernel<<<1536, 256, 0, stream>>>(
      (const float4*)ck, (const float4*)cv, (float4*)kout, (float4*)vout);

  // 2) projections (q -> ws; k_new/v_new -> output rows [3072,4096))
  proj_kernel<<<1536, 128, 0, stream>>>(x, Wq, Wk, Wv, qws, kout, vout);

  // 3) flash attention over the completed k/v
  attn_kernel<<<256, 64, 0, stream>>>(qws, kout, vout, out);
}